// VolumeSampler_76897094468377
// MI455X (gfx1250) — compile-verified
//
#include <hip/hip_runtime.h>
#include <hip/hip_bf16.h>
#include <math.h>
#include <stdint.h>

// ---------------------------------------------------------------------------
// VolumeSampler for MI455X (gfx1250).
// Pipeline:
//   K1 zero_ws      : clear 10*V float accumulators in d_ws (40 MB, L2-resident)
//   K2 scatter      : stream points through LDS with double-buffered
//                     global_load_async_to_lds_b128 (ASYNCcnt pipeline),
//                     atomic-accumulate {count, sum(d), sum(d d^T)} with
//                     d = x - voxel_center  (well-conditioned moments)
//   K3 bcenter      : downsampled voxel centers (131072 x float4)
//   K4 eig_val      : closed-form 3x3 symmetric eigenvalues (ascending)
//   K5 eig_vec      : eigenvectors via (A-l1 I)(A-l2 I) products, batched
//                     5 voxels per V_WMMA_F32_16X16X4_F32 (block-diagonal
//                     3x3 blocks in the 16x16 tile), wave32, 15 voxels/wave.
// ---------------------------------------------------------------------------

#define B_   4
#define GX_  128
#define GY_  128
#define GZ_  16
#define V_   (B_ * GX_ * GY_ * GZ_)               // 1,048,576
#define EPS_ 1.0e-4f

#define NBC_           (B_ * (GX_/2) * (GY_/2) * (GZ_/2))   // 131072
#define OUT_EVAL_OFF   ((size_t)(NBC_ * 4))                 // 524,288 floats
#define OUT_EVEC_OFF   (OUT_EVAL_OFF + 3ull * (size_t)V_)

typedef float v2f __attribute__((ext_vector_type(2)));
typedef float v8f __attribute__((ext_vector_type(8)));

#if __has_builtin(__builtin_amdgcn_wmma_f32_16x16x4_f32)
#define HAVE_WMMA_F32X4 1
#endif

// ---------------------------------------------------------------------------
__global__ __launch_bounds__(256) void zero_ws_kernel(float4* __restrict__ w, int n4) {
  const float4 z = make_float4(0.f, 0.f, 0.f, 0.f);
  for (int i = blockIdx.x * blockDim.x + threadIdx.x; i < n4;
       i += gridDim.x * blockDim.x)
    w[i] = z;
}

// ---------------------------------------------------------------------------
// Scatter: double-buffered async global->LDS point staging.
#define TILE 256

__global__ __launch_bounds__(256) void scatter_kernel(const float4* __restrict__ pts,
                                                      const float* __restrict__ vsz,
                                                      float* __restrict__ ws, int n) {
  __shared__ float4 tile[2][TILE];
  const float vx = vsz[0], vy = vsz[1], vz = vsz[2];
  const int tid = threadIdx.x;
  const int ntiles = (n + TILE - 1) / TILE;

  int t = (int)blockIdx.x;
  int buf = 0;
  if (t < ntiles) {
    const int idx = t * TILE + tid;
    if (idx < n) {
      const unsigned long long g = (unsigned long long)(uintptr_t)(pts + idx);
      const unsigned lds = (unsigned)(uintptr_t)&tile[0][tid];
      asm volatile("global_load_async_to_lds_b128 %0, %1, off"
                   :: "v"(lds), "v"(g) : "memory");
    }
  }
  for (; t < ntiles; t += (int)gridDim.x) {
    const int tn = t + (int)gridDim.x;
    if (tn < ntiles) {
      const int idx = tn * TILE + tid;
      if (idx < n) {
        const unsigned long long g = (unsigned long long)(uintptr_t)(pts + idx);
        const unsigned lds = (unsigned)(uintptr_t)&tile[buf ^ 1][tid];
        asm volatile("global_load_async_to_lds_b128 %0, %1, off"
                     :: "v"(lds), "v"(g) : "memory");
      }
      asm volatile("s_wait_asynccnt 1" ::: "memory");   // current tile landed
    } else {
      asm volatile("s_wait_asynccnt 0" ::: "memory");
    }

    const int idx = t * TILE + tid;
    if (idx < n) {
      const float4 p = tile[buf][tid];
      const int b = (int)p.x;
      int cx = (int)floorf(p.y / vx); cx = cx < 0 ? 0 : (cx > GX_-1 ? GX_-1 : cx);
      int cy = (int)floorf(p.z / vy); cy = cy < 0 ? 0 : (cy > GY_-1 ? GY_-1 : cy);
      int cz = (int)floorf(p.w / vz); cz = cz < 0 ? 0 : (cz > GZ_-1 ? GZ_-1 : cz);
      const int v = ((b * GX_ + cx) * GY_ + cy) * GZ_ + cz;
      // center-relative moments: |d| <= 0.4 -> no cancellation in cov
      const float dx = p.y - ((float)cx + 0.5f) * vx;
      const float dy = p.z - ((float)cy + 0.5f) * vy;
      const float dz = p.w - ((float)cz + 0.5f) * vz;
      atomicAdd(&ws[0*V_ + v], 1.0f);
      atomicAdd(&ws[1*V_ + v], dx);
      atomicAdd(&ws[2*V_ + v], dy);
      atomicAdd(&ws[3*V_ + v], dz);
      atomicAdd(&ws[4*V_ + v], dx*dx);
      atomicAdd(&ws[5*V_ + v], dx*dy);
      atomicAdd(&ws[6*V_ + v], dx*dz);
      atomicAdd(&ws[7*V_ + v], dy*dy);
      atomicAdd(&ws[8*V_ + v], dy*dz);
      atomicAdd(&ws[9*V_ + v], dz*dz);
    }
    buf ^= 1;
  }
}

// ---------------------------------------------------------------------------
__global__ __launch_bounds__(256) void bcenter_kernel(const float* __restrict__ vsz,
                                                      float* __restrict__ out) {
  const int j = blockIdx.x * blockDim.x + threadIdx.x;
  if (j >= NBC_) return;
  const int b   = j >> 15;            // / (64*64*8)
  const int rem = j & 32767;
  const int gx = (rem >> 9) << 1;     // / (64*8), * DS
  const int gy = ((rem >> 3) & 63) << 1;
  const int gz = (rem & 7) << 1;
  float4 o;
  o.x = (float)b;
  o.y = ((float)gx + 0.5f) * vsz[0];
  o.z = ((float)gy + 0.5f) * vsz[1];
  o.w = ((float)gz + 0.5f) * vsz[2];
  ((float4*)out)[j] = o;
}

// ---------------------------------------------------------------------------
__device__ __forceinline__ void load_cov(const float* __restrict__ ws, int v,
                                         float& a00, float& a01, float& a02,
                                         float& a11, float& a12, float& a22) {
  const float cnt = ws[v];
  const float vol = fmaxf(cnt, 1.0f);
  const float inv = 1.0f / vol;
  const float mx = ws[1*V_ + v] * inv;
  const float my = ws[2*V_ + v] * inv;
  const float mz = ws[3*V_ + v] * inv;
  // empty voxel: all sums 0 -> pure EPS*diag(1,2,3), matches reference
  a00 = fmaf(-mx, mx, ws[4*V_ + v] * inv) + EPS_ * 1.0f;
  a01 = fmaf(-mx, my, ws[5*V_ + v] * inv);
  a02 = fmaf(-mx, mz, ws[6*V_ + v] * inv);
  a11 = fmaf(-my, my, ws[7*V_ + v] * inv) + EPS_ * 2.0f;
  a12 = fmaf(-my, mz, ws[8*V_ + v] * inv);
  a22 = fmaf(-mz, mz, ws[9*V_ + v] * inv) + EPS_ * 3.0f;
}

__device__ __forceinline__ void sym3_eigvals(float a00, float a01, float a02,
                                             float a11, float a12, float a22,
                                             float& l0, float& l1, float& l2) {
  const float q  = (a00 + a11 + a22) * (1.0f/3.0f);
  const float p1 = a01*a01 + a02*a02 + a12*a12;
  const float b00 = a00 - q, b11 = a11 - q, b22 = a22 - q;
  const float p2 = b00*b00 + b11*b11 + b22*b22 + 2.0f*p1;
  if (p2 < 1.0e-22f) { l0 = q; l1 = q; l2 = q; return; }
  const float p  = sqrtf(p2 * (1.0f/6.0f));
  const float ip = 1.0f / p;
  const float c00 = b00*ip, c01 = a01*ip, c02 = a02*ip;
  const float c11 = b11*ip, c12 = a12*ip, c22 = b22*ip;
  float r = 0.5f * (c00*(c11*c22 - c12*c12)
                  - c01*(c01*c22 - c12*c02)
                  + c02*(c01*c12 - c11*c02));
  r = fminf(fmaxf(r, -1.0f), 1.0f);
  const float phi = acosf(r) * (1.0f/3.0f);
  l2 = q + 2.0f*p*cosf(phi);
  l0 = q + 2.0f*p*cosf(phi + 2.0943951023931953f);  // +2pi/3
  l1 = 3.0f*q - l0 - l2;
}

__global__ __launch_bounds__(256) void eig_val_kernel(const float* __restrict__ ws,
                                                      float* __restrict__ out) {
  const int v = blockIdx.x * blockDim.x + threadIdx.x;
  if (v >= V_) return;
  float a00, a01, a02, a11, a12, a22;
  load_cov(ws, v, a00, a01, a02, a11, a12, a22);
  float l0, l1, l2;
  sym3_eigvals(a00, a01, a02, a11, a12, a22, l0, l1, l2);
  const size_t base = OUT_EVAL_OFF + 3ull * (size_t)v;
  out[base + 0] = l0; out[base + 1] = l1; out[base + 2] = l2;
}

// ---------------------------------------------------------------------------
// Eigenvectors. One wave (32 threads) per block; 15 voxels per iteration.
// Per voxel we need two 3x3 products P0=(A-l1)(A-l2), P2=(A-l0)(A-l1);
// WMMA f32 16x16x4 computes 5 of them block-diagonally per instruction.

#define EVQ 15

__device__ __forceinline__ int symidx(int r, int c) {
  if (r > c) { const int t = r; r = c; c = t; }
  return (r == 0) ? c : ((r == 1) ? 2 + c : 5);
}

__global__ __launch_bounds__(32) void eig_vec_kernel(const float* __restrict__ ws,
                                                     float* __restrict__ out,
                                                     int ngroups) {
  __shared__ float sA[EVQ][6];
  __shared__ float sL[EVQ][3];
  __shared__ float sD[16][16];
  const int lane = threadIdx.x;

  for (int g = blockIdx.x; g < ngroups; g += gridDim.x) {
    const int myv = g * EVQ + lane;
    const bool valid = (lane < EVQ) && (myv < V_);
    if (lane < EVQ) {
      if (myv < V_) {
        float a00, a01, a02, a11, a12, a22;
        load_cov(ws, myv, a00, a01, a02, a11, a12, a22);
        sA[lane][0] = a00; sA[lane][1] = a01; sA[lane][2] = a02;
        sA[lane][3] = a11; sA[lane][4] = a12; sA[lane][5] = a22;
        const size_t eb = OUT_EVAL_OFF + 3ull * (size_t)myv;
        sL[lane][0] = out[eb+0]; sL[lane][1] = out[eb+1]; sL[lane][2] = out[eb+2];
      } else {   // pad slot: harmless distinct-eigenvalue diagonal
        sA[lane][0] = EPS_;   sA[lane][1] = 0.f; sA[lane][2] = 0.f;
        sA[lane][3] = 2*EPS_; sA[lane][4] = 0.f; sA[lane][5] = 3*EPS_;
        sL[lane][0] = EPS_;   sL[lane][1] = 2*EPS_; sL[lane][2] = 3*EPS_;
      }
    }
    __syncthreads();

    float P0[3][3] = {{0}}, P2[3][3] = {{0}};
#ifdef HAVE_WMMA_F32X4
    {
      // A 16x4 f32 layout: vgpr j -> (M = lane&15, K = j + (lane<16 ? 0:2))
      // B 4x16 f32 layout: vgpr j -> (K = j + (lane<16 ? 0:2), N = lane&15)
      // D 16x16 f32 layout: vgpr r -> (M = r + (lane<16 ? 0:8), N = lane&15)
      const int M   = lane & 15;
      const int kb  = (lane < 16) ? 0 : 2;
      const int blk = (M < 15) ? (M / 3) : 0;   // 0..4 within a batch
      const int r   = M % 3;
      #pragma unroll
      for (int bb = 0; bb < 3; ++bb) {          // 3 batches of 5 voxels
        const int s = 5 * bb + blk;             // shared-mem voxel slot 0..14
        #pragma unroll
        for (int which = 0; which < 2; ++which) {
          const float lamL = (which == 0) ? sL[s][1] : sL[s][0];
          const float lamR = (which == 0) ? sL[s][2] : sL[s][1];
          v2f av, bv;
          #pragma unroll
          for (int jj = 0; jj < 2; ++jj) {
            const int K = kb + jj;
            float aa = 0.0f, bbv = 0.0f;
            if (K < 3 && M < 15) {
              aa  = sA[s][symidx(r, K)] - ((r == K) ? lamL : 0.0f);
              bbv = sA[s][symidx(K, r)] - ((K == r) ? lamR : 0.0f);
            }
            av[jj] = aa; bv[jj] = bbv;
          }
          const v8f cz = {0.f,0.f,0.f,0.f,0.f,0.f,0.f,0.f};
          const v8f d = __builtin_amdgcn_wmma_f32_16x16x4_f32(
              false, av, false, bv, (short)0, cz, false, false);
          __syncthreads();
          #pragma unroll
          for (int rr = 0; rr < 8; ++rr)
            sD[(lane < 16) ? rr : rr + 8][lane & 15] = d[rr];
          __syncthreads();
          const int li = lane - 5 * bb;         // my block within this batch
          if (li >= 0 && li < 5) {
            #pragma unroll
            for (int rr = 0; rr < 3; ++rr)
              #pragma unroll
              for (int cc = 0; cc < 3; ++cc) {
                const float val = sD[3*li + rr][3*li + cc];
                if (which == 0) P0[rr][cc] = val; else P2[rr][cc] = val;
              }
          }
          __syncthreads();
        }
      }
    }
#else
    if (lane < EVQ) {
      const float Am[3][3] = {{sA[lane][0], sA[lane][1], sA[lane][2]},
                              {sA[lane][1], sA[lane][3], sA[lane][4]},
                              {sA[lane][2], sA[lane][4], sA[lane][5]}};
      const float l0 = sL[lane][0], l1 = sL[lane][1], l2 = sL[lane][2];
      #pragma unroll
      for (int rr = 0; rr < 3; ++rr)
        #pragma unroll
        for (int cc = 0; cc < 3; ++cc) {
          float acc0 = 0.f, acc2 = 0.f;
          #pragma unroll
          for (int k = 0; k < 3; ++k) {
            acc0 += (Am[rr][k] - ((rr==k)?l1:0.f)) * (Am[k][cc] - ((k==cc)?l2:0.f));
            acc2 += (Am[rr][k] - ((rr==k)?l0:0.f)) * (Am[k][cc] - ((k==cc)?l1:0.f));
          }
          P0[rr][cc] = acc0; P2[rr][cc] = acc2;
        }
    }
    __syncthreads();
#endif

    if (valid) {
      float w0[3], w1[3], w2[3];
      // largest-norm column, constant indexing (no scratch spills)
      {
        const float q0 = P2[0][0]*P2[0][0] + P2[1][0]*P2[1][0] + P2[2][0]*P2[2][0];
        const float q1 = P2[0][1]*P2[0][1] + P2[1][1]*P2[1][1] + P2[2][1]*P2[2][1];
        const float q2 = P2[0][2]*P2[0][2] + P2[1][2]*P2[1][2] + P2[2][2]*P2[2][2];
        int bc = 0; float best = q0;
        if (q1 > best) { best = q1; bc = 1; }
        if (q2 > best) { best = q2; bc = 2; }
        if (best > 1e-26f) {
          const float sI = rsqrtf(best);
          w2[0] = (bc==0?P2[0][0]:(bc==1?P2[0][1]:P2[0][2])) * sI;
          w2[1] = (bc==0?P2[1][0]:(bc==1?P2[1][1]:P2[1][2])) * sI;
          w2[2] = (bc==0?P2[2][0]:(bc==1?P2[2][1]:P2[2][2])) * sI;
        } else { w2[0] = 0.f; w2[1] = 0.f; w2[2] = 1.f; }
      }
      {
        const float q0 = P0[0][0]*P0[0][0] + P0[1][0]*P0[1][0] + P0[2][0]*P0[2][0];
        const float q1 = P0[0][1]*P0[0][1] + P0[1][1]*P0[1][1] + P0[2][1]*P0[2][1];
        const float q2 = P0[0][2]*P0[0][2] + P0[1][2]*P0[1][2] + P0[2][2]*P0[2][2];
        int bc = 0; float best = q0;
        if (q1 > best) { best = q1; bc = 1; }
        if (q2 > best) { best = q2; bc = 2; }
        if (best > 1e-26f) {
          const float sI = rsqrtf(best);
          w0[0] = (bc==0?P0[0][0]:(bc==1?P0[0][1]:P0[0][2])) * sI;
          w0[1] = (bc==0?P0[1][0]:(bc==1?P0[1][1]:P0[1][2])) * sI;
          w0[2] = (bc==0?P0[2][0]:(bc==1?P0[2][1]:P0[2][2])) * sI;
        } else { w0[0] = 1.f; w0[1] = 0.f; w0[2] = 0.f; }
      }
      // Gram-Schmidt w0 against w2 (degenerate-eigenvalue safety)
      float dp = w0[0]*w2[0] + w0[1]*w2[1] + w0[2]*w2[2];
      w0[0] -= dp*w2[0]; w0[1] -= dp*w2[1]; w0[2] -= dp*w2[2];
      float nn = w0[0]*w0[0] + w0[1]*w0[1] + w0[2]*w0[2];
      if (nn < 1e-12f) {
        const float ax = fabsf(w2[0]), ay = fabsf(w2[1]), az = fabsf(w2[2]);
        float e0 = 0.f, e1 = 0.f, e2 = 0.f;
        if (ax <= ay && ax <= az) e0 = 1.f; else if (ay <= az) e1 = 1.f; else e2 = 1.f;
        dp = e0*w2[0] + e1*w2[1] + e2*w2[2];
        w0[0] = e0 - dp*w2[0]; w0[1] = e1 - dp*w2[1]; w0[2] = e2 - dp*w2[2];
        nn = w0[0]*w0[0] + w0[1]*w0[1] + w0[2]*w0[2];
      }
      const float s0 = rsqrtf(fmaxf(nn, 1e-30f));
      w0[0] *= s0; w0[1] *= s0; w0[2] *= s0;
      w1[0] = w2[1]*w0[2] - w2[2]*w0[1];
      w1[1] = w2[2]*w0[0] - w2[0]*w0[2];
      w1[2] = w2[0]*w0[1] - w2[1]*w0[0];

      const size_t base = OUT_EVEC_OFF + 9ull * (size_t)myv;
      out[base+0] = w0[0]; out[base+1] = w1[0]; out[base+2] = w2[0];
      out[base+3] = w0[1]; out[base+4] = w1[1]; out[base+5] = w2[1];
      out[base+6] = w0[2]; out[base+7] = w1[2]; out[base+8] = w2[2];
    }
    __syncthreads();
  }
}

// ---------------------------------------------------------------------------
extern "C" void kernel_launch(void* const* d_in, const int* in_sizes, int n_in,
                              void* d_out, int out_size, void* d_ws, size_t ws_size,
                              hipStream_t stream) {
  (void)n_in; (void)out_size; (void)ws_size;
  const float4* pts = (const float4*)d_in[0];   // point_bxyz, N x 4 f32
  const float*  vsz = (const float*)d_in[1];    // voxel_size, 3 f32
  float* out = (float*)d_out;
  float* ws  = (float*)d_ws;                    // 10*V floats (40 MB)
  const int n = in_sizes[0] / 4;

  zero_ws_kernel<<<4096, 256, 0, stream>>>((float4*)d_ws, (10 * V_) / 4);
  scatter_kernel<<<1024, 256, 0, stream>>>(pts, vsz, ws, n);
  bcenter_kernel<<<(NBC_ + 255) / 256, 256, 0, stream>>>(vsz, out);
  eig_val_kernel<<<(V_ + 255) / 256, 256, 0, stream>>>(ws, out);
  const int ngroups = (V_ + EVQ - 1) / EVQ;
  eig_vec_kernel<<<ngroups, 32, 0, stream>>>(ws, out, ngroups);
}